// QuantumLayer_64501818851435
// MI455X (gfx1250) — compile-verified
//
#include <hip/hip_runtime.h>

typedef __attribute__((ext_vector_type(2))) float v2f;
typedef __attribute__((ext_vector_type(8))) float v8f;

#define N_QUBITS 12
#define N_LAYERS 6
#define BATCH    4096
#define DIM      4096   // 2^12
#define NU       (N_LAYERS * 3)        // 18 fused 16x16 unitaries
#define UFLOATS  (NU * 512)            // 9216 floats (re+im per unitary)

static __device__ __forceinline__ v8f wmma4(v2f a, v2f b, v8f c) {
  // D(16x16,f32) += A(16x4,f32) * B(4x16,f32)   [+ C]
  return __builtin_amdgcn_wmma_f32_16x16x4_f32(false, a, false, b, (short)0, c,
                                               false, false);
}

// ---------------------------------------------------------------------------
// Kernel A: fuse each layer's per-qubit RZ(tz)*RY(ty) gates into three 16x16
// complex unitaries (Kronecker product over wire groups {0-3},{4-7},{8-11}).
// Umats[(l*3+g)*512 + 0..255] = Re(U), [256..511] = Im(U), row-major 16x16.
// U[i][j] = prod_k G_{4g+k}[bit(3-k,i)][bit(3-k,j)],  G = RZ * RY.
// ---------------------------------------------------------------------------
__global__ __launch_bounds__(256) void fuse_unitaries(
    const float* __restrict__ weights, float* __restrict__ Umats) {
  const int e = blockIdx.x;          // 0..17  (= l*3 + g)
  const int l = e / 3, g = e % 3;
  const int tid = threadIdx.x;       // 256 threads = 16x16 entries
  const int i = tid >> 4, j = tid & 15;

  float pr = 1.f, pi = 0.f;
#pragma unroll
  for (int k = 0; k < 4; ++k) {
    const int q  = 4 * g + k;
    const float ty = 0.5f * weights[(l * N_QUBITS + q) * 2 + 0];
    const float tz = 0.5f * weights[(l * N_QUBITS + q) * 2 + 1];
    const float cy = cosf(ty), sy = sinf(ty);
    const float cz = cosf(tz), sz = sinf(tz);
    const int bi = (i >> (3 - k)) & 1;
    const int bj = (j >> (3 - k)) & 1;
    const float ryv = bi ? (bj ? cy : sy) : (bj ? -sy : cy);
    const float gre = ryv * cz;
    const float gim = ryv * (bi ? sz : -sz);
    const float nr = pr * gre - pi * gim;
    const float ni = pr * gim + pi * gre;
    pr = nr; pi = ni;
  }
  Umats[e * 512 + tid]       = pr;
  Umats[e * 512 + 256 + tid] = pi;
}

// ---------------------------------------------------------------------------
// Kernel B: one workgroup (256 threads = 8 wave32) per batch sample.
// State in LDS (Sr/Si, 32 KB). Per layer: three fused 16x16 complex unitaries
// applied via V_WMMA_F32_16X16X4_F32.  The inter-layer CNOT ladder
// (CNOT(i,i+1), i=0..10) is a pure basis relabeling whose inverse is the
// Gray code a ^ (a>>1); it is folded into the next layer's stage-A reads,
// so no swaps or extra barriers are ever executed.
// ---------------------------------------------------------------------------
__global__ __launch_bounds__(256) void qsim(const float* __restrict__ x,
                                            const float* __restrict__ Umats,
                                            float* __restrict__ out) {
  __shared__ float Sr[DIM];
  __shared__ float Si[DIM];
  __shared__ float Uall[UFLOATS];    // all 18 unitaries, loaded once (36 KB)
  __shared__ float red[256];

  const int tid  = threadIdx.x;
  const int b    = blockIdx.x;
  const int lane = tid & 31;
  const int wave = tid >> 5;      // 0..7
  const int half = lane >> 4;     // 0/1
  const int lm   = lane & 15;     // 0..15

  // ---- preload all fused unitaries (9216 floats / 256 threads = 36 each)
#pragma unroll
  for (int k = 0; k < UFLOATS / 256; ++k)
    Uall[tid + 256 * k] = Umats[tid + 256 * k];

  // ---- AngleEmbedding on |0..0> collapses to a product state:
  //      amp[a] = prod_w ( bit(11-w,a) ? sin(x_w/2) : cos(x_w/2) )
  float cw[N_QUBITS], sw[N_QUBITS];
#pragma unroll
  for (int w = 0; w < N_QUBITS; ++w) {
    const float t = 0.5f * x[b * N_QUBITS + w];
    cw[w] = cosf(t);
    sw[w] = sinf(t);
  }
#pragma unroll
  for (int k = 0; k < 16; ++k) {
    const int a = tid + 256 * k;
    float p = 1.f;
#pragma unroll
    for (int w = 0; w < N_QUBITS; ++w)
      p *= ((a >> (11 - w)) & 1) ? sw[w] : cw[w];
    Sr[a] = p;
    Si[a] = 0.f;
  }
  __syncthreads();

  for (int l = 0; l < N_LAYERS; ++l) {
    const bool gray = (l > 0);     // pending CNOT-ladder permutation to fold
    for (int g = 0; g < 3; ++g) {
      const float* Ur = Uall + (l * 3 + g) * 512;
      const float* Ui = Ur + 256;

      v8f dr[2] = {};
      v8f di[2] = {};
#pragma unroll
      for (int c = 0; c < 4; ++c) {          // K chunks of 4
        const int k0 = 4 * c + 2 * half;     // K of operand VGPR0 (VGPR1=k0+1)
        v2f ur, ui, nui;
        ur.x = Ur[lm * 16 + k0];
        ur.y = Ur[lm * 16 + k0 + 1];
        ui.x = Ui[lm * 16 + k0];
        ui.y = Ui[lm * 16 + k0 + 1];
        nui  = -ui;                          // f32 WMMA has no A/B NEG modifier
#pragma unroll
        for (int t2 = 0; t2 < 2; ++t2) {
          const int tile = wave * 2 + t2;    // 16 tiles per stage, 2 per wave
          int a0, a1;
          if (g == 0) {        // wires 0-3:  S' = U * S   (S as 16 x 256)
            a0 = k0 * 256 + tile * 16 + lm;
            a1 = a0 + 256;
            if (gray) {        // fold inverse CNOT-ladder perm (Gray code)
              a0 ^= (a0 >> 1);
              a1 ^= (a1 >> 1);
            }
          } else if (g == 1) { // wires 4-7:  batched 16x16 GEMMs over hi=tile
            a0 = tile * 256 + k0 * 16 + lm;  a1 = a0 + 16;
          } else {             // wires 8-11: S' = S * U^T (S as 256 x 16)
            a0 = (tile * 16 + lm) * 16 + k0; a1 = a0 + 1;
          }
          v2f br, bi;
          br.x = Sr[a0]; br.y = Sr[a1];
          bi.x = Si[a0]; bi.y = Si[a1];
          if (g < 2) {
            // D = U * S : Dr = Ur*Sr - Ui*Si ; Di = Ur*Si + Ui*Sr
            dr[t2] = wmma4(ur,  br, dr[t2]);
            dr[t2] = wmma4(nui, bi, dr[t2]);
            di[t2] = wmma4(ur,  bi, di[t2]);
            di[t2] = wmma4(ui,  br, di[t2]);
          } else {
            // D = S * U^T : Dr = Sr*Ur' - Si*Ui' ; Di = Sr*Ui' + Si*Ur'
            dr[t2] = wmma4(br, ur,  dr[t2]);
            dr[t2] = wmma4(bi, nui, dr[t2]);
            di[t2] = wmma4(br, ui,  di[t2]);
            di[t2] = wmma4(bi, ur,  di[t2]);
          }
        }
      }
      __syncthreads();   // all waves done reading Sr/Si
#pragma unroll
      for (int t2 = 0; t2 < 2; ++t2) {
        const int tile = wave * 2 + t2;
#pragma unroll
        for (int v = 0; v < 8; ++v) {
          const int m = v + 8 * half;        // D layout: M = vgpr + 8*(lane/16)
          int a;
          if (g == 0)      a = m * 256 + tile * 16 + lm;
          else if (g == 1) a = tile * 256 + m * 16 + lm;
          else             a = (tile * 16 + m) * 16 + lm;
          Sr[a] = dr[t2][v];
          Si[a] = di[t2][v];
        }
      }
      __syncthreads();
    }
  }

  // ---- <Z_0> : wire 0 is the MSB (bit 11) of the amplitude index
  float acc = 0.f;
#pragma unroll
  for (int k = 0; k < 16; ++k) {
    const int a = tid + 256 * k;
    const float p = Sr[a] * Sr[a] + Si[a] * Si[a];
    acc += (a & 2048) ? -p : p;
  }
  red[tid] = acc;
  __syncthreads();
  for (int off = 128; off > 0; off >>= 1) {
    if (tid < off) red[tid] += red[tid + off];
    __syncthreads();
  }
  if (tid == 0) out[b] = red[0];
}

// ---------------------------------------------------------------------------
extern "C" void kernel_launch(void* const* d_in, const int* in_sizes, int n_in,
                              void* d_out, int out_size, void* d_ws,
                              size_t ws_size, hipStream_t stream) {
  const float* x = (const float*)d_in[0];      // (4096, 12) f32
  const float* w = (const float*)d_in[1];      // (6, 12, 2) f32
  float* out     = (float*)d_out;              // (4096,)    f32
  float* Umats   = (float*)d_ws;               // 18*512 floats = 36 KB scratch

  fuse_unitaries<<<dim3(N_LAYERS * 3), dim3(256), 0, stream>>>(w, Umats);
  qsim<<<dim3(BATCH), dim3(256), 0, stream>>>(x, Umats, out);
}